// StateNeuron_489626272326
// MI455X (gfx1250) — compile-verified
//
#include <hip/hip_runtime.h>
#include <math.h>

// ---------------------------------------------------------------------------
// CDNA5 (gfx1250) chunked linear-scan SSM:
//   y_t = tanh(C h_t + D u_t + n_t),  h_{t+1} = A h_t + B u_t,  h_0 = ones
// ORDER = IN = OUT = 32, SEQ = 524288.
// All heavy matmuls are 32x32 @ 32x16 tiles on V_WMMA_F32_16X16X4_F32 (fp32).
// ---------------------------------------------------------------------------

#define SEQ     524288
#define L       64                 // time steps per chunk
#define CHUNKS  (SEQ / L)          // 8192
#define GC      16                 // chunk columns batched per wave (WMMA N)
#define NG      (CHUNKS / GC)      // 512 wave-groups for phases 1 & 3
#define SUPER   64                 // chunks per super-chunk
#define NSUP    (CHUNKS / SUPER)   // 128 super-chunks
#define NSG     (NSUP / GC)        // 8 wave-groups for phase-2 batched scans

typedef __attribute__((ext_vector_type(2))) float v2f;
typedef __attribute__((ext_vector_type(8))) float v8f;

// --- fragment helpers ------------------------------------------------------
// A-operand (16x4 f32) layout per CDNA5 ISA 7.12.2:
//   lanes 0-15 : row = lane,    v0 = K0, v1 = K1
//   lanes 16-31: row = lane-16, v0 = K2, v1 = K3
// B-operand (4x16 f32) mirrored: lanes hold column, VGPRs hold K pairs.
// C/D (16x16 f32): VGPR v -> row v (lanes 0-15) / row v+8 (lanes 16-31),
//                  column = lane & 15.

__device__ __forceinline__ void zacc(v8f a[2]) {
  for (int r = 0; r < 2; ++r)
    for (int e = 0; e < 8; ++e) a[r][e] = 0.0f;
}

// Preload loop-invariant A-operand fragments of a 32x32 row-major matrix W.
// f[rowTile][k4] covers output rows rowTile*16.. and K-slice k4*4..
__device__ __forceinline__ void preload(const float* W, float2 f[2][8]) {
  const int lane = threadIdx.x & 31;
  const int row  = lane & 15;
  const int ksub = (lane >> 4) << 1;           // 0 for lanes<16, 2 otherwise
  for (int r = 0; r < 2; ++r)
    for (int k4 = 0; k4 < 8; ++k4) {
      const int rr = r * 16 + row;
      const int kk = k4 * 4 + ksub;
      f[r][k4] = make_float2(W[rr * 32 + kk], W[rr * 32 + kk + 1]);
    }
}

// acc(32x16) += W(32x32) @ X(32x16); W given as preloaded fragments,
// X as a 32x16 row-major LDS tile.
__device__ __forceinline__ void gemm_acc(const float2 f[2][8],
                                         const float* X, v8f acc[2]) {
  const int lane = threadIdx.x & 31;
  const int col  = lane & 15;
  const int ksub = (lane >> 4) << 1;
  for (int k4 = 0; k4 < 8; ++k4) {
    const int kk = k4 * 4 + ksub;
    v2f b;
    b.x = X[kk * 16 + col];
    b.y = X[(kk + 1) * 16 + col];
    for (int r = 0; r < 2; ++r) {
      v2f a;
      a.x = f[r][k4].x;
      a.y = f[r][k4].y;
      acc[r] = __builtin_amdgcn_wmma_f32_16x16x4_f32(
          /*neg_a=*/false, a, /*neg_b=*/false, b,
          /*c_mod=*/(short)0, acc[r], /*reuse_a=*/false, /*reuse_b=*/false);
    }
  }
}

// Store 32x16 accumulator pair into a row-major LDS tile.
__device__ __forceinline__ void store_acc(float* X, const v8f acc[2]) {
  const int lane  = threadIdx.x & 31;
  const int col   = lane & 15;
  const int mBase = (lane >> 4) << 3;          // lanes>=16 hold rows v+8
  for (int r = 0; r < 2; ++r)
    for (int e = 0; e < 8; ++e)
      X[(r * 16 + mBase + e) * 16 + col] = acc[r][e];
}

// acc += T (32x16 row-major LDS tile), mapped through C/D layout.
__device__ __forceinline__ void add_tile(v8f acc[2], const float* T) {
  const int lane  = threadIdx.x & 31;
  const int col   = lane & 15;
  const int mBase = (lane >> 4) << 3;
  for (int r = 0; r < 2; ++r)
    for (int e = 0; e < 8; ++e)
      acc[r][e] += T[(r * 16 + mBase + e) * 16 + col];
}

// --- Phase 1: per-chunk local scans with h0 = 0 -> S_c ---------------------
__global__ __launch_bounds__(32) void k_phase1(const float* __restrict__ u,
                                               const float* __restrict__ A,
                                               const float* __restrict__ B,
                                               float* __restrict__ S) {
  __shared__ float sA[1024], sB[1024], hT[512], uT[512];
  const int lane = threadIdx.x;
  const int g = blockIdx.x;
  for (int i = lane; i < 1024; i += 32) { sA[i] = A[i]; sB[i] = B[i]; }
  for (int i = lane; i < 512; i += 32) hT[i] = 0.0f;
  __syncthreads();
  float2 fA[2][8], fB[2][8];
  preload(sA, fA);
  preload(sB, fB);
  const int tBase = g * GC * L;
  for (int j = 0; j < L; ++j) {
    for (int c = 0; c < GC; ++c)
      uT[lane * GC + c] = u[lane * SEQ + tBase + c * L + j];
    __syncthreads();
    v8f acc[2];
    zacc(acc);
    gemm_acc(fA, hT, acc);     // A @ h
    gemm_acc(fB, uT, acc);     // + B @ u
    __syncthreads();
    store_acc(hT, acc);
    __syncthreads();
  }
  for (int c = 0; c < GC; ++c)
    S[(g * GC + c) * 32 + lane] = hT[lane * GC + c];
}

// --- Matrix powers: P = A^64, P2 = A^4096 (tiny, one wave) -----------------
__global__ __launch_bounds__(32) void k_powers(const float* __restrict__ A,
                                               float* __restrict__ P,
                                               float* __restrict__ P2) {
  __shared__ float M[1024], T[1024];
  const int lane = threadIdx.x;
  for (int i = lane; i < 1024; i += 32) M[i] = A[i];
  __syncthreads();
  for (int s = 0; s < 12; ++s) {             // 12 squarings total
    for (int n = 0; n < 32; ++n) {
      float acc = 0.0f;
      for (int k = 0; k < 32; ++k) acc += M[lane * 32 + k] * M[k * 32 + n];
      T[lane * 32 + n] = acc;
    }
    __syncthreads();
    for (int i = lane; i < 1024; i += 32) M[i] = T[i];
    __syncthreads();
    if (s == 5)                              // after 6 squarings: A^64
      for (int i = lane; i < 1024; i += 32) P[i] = M[i];
    __syncthreads();
  }
  for (int i = lane; i < 1024; i += 32) P2[i] = M[i];   // A^4096
}

// --- Phase 2a: per-super-chunk local scans (h'=0): S2_s --------------------
__global__ __launch_bounds__(32) void k_p2a(const float* __restrict__ S,
                                            const float* __restrict__ P,
                                            float* __restrict__ S2) {
  __shared__ float sP[1024], hT[512], sT[512];
  const int lane = threadIdx.x;
  const int g = blockIdx.x;
  for (int i = lane; i < 1024; i += 32) sP[i] = P[i];
  for (int i = lane; i < 512; i += 32) hT[i] = 0.0f;
  __syncthreads();
  float2 fP[2][8];
  preload(sP, fP);
  for (int m = 0; m < SUPER; ++m) {
    for (int c = 0; c < GC; ++c)
      sT[lane * GC + c] = S[((g * GC + c) * SUPER + m) * 32 + lane];
    __syncthreads();
    v8f acc[2];
    zacc(acc);
    gemm_acc(fP, hT, acc);     // P @ h'
    add_tile(acc, sT);         // + S_c
    __syncthreads();
    store_acc(hT, acc);
    __syncthreads();
  }
  for (int c = 0; c < GC; ++c)
    S2[(g * GC + c) * 32 + lane] = hT[lane * GC + c];
}

// --- Phase 2b: top-level sequential scan over 128 super-chunks -------------
__global__ __launch_bounds__(32) void k_p2b(const float* __restrict__ S2,
                                            const float* __restrict__ P2m,
                                            float* __restrict__ G) {
  __shared__ float h[32];
  const int lane = threadIdx.x;
  float prow[32];
  for (int k = 0; k < 32; ++k) prow[k] = P2m[lane * 32 + k];
  h[lane] = 1.0f;                             // h_0 = ones
  __syncthreads();
  for (int s = 0; s < NSUP; ++s) {
    G[s * 32 + lane] = h[lane];
    float acc = S2[s * 32 + lane];
    for (int k = 0; k < 32; ++k) acc += prow[k] * h[k];
    __syncthreads();
    h[lane] = acc;
    __syncthreads();
  }
}

// --- Phase 2c: fixup -> initial state H_c for every chunk ------------------
__global__ __launch_bounds__(32) void k_p2c(const float* __restrict__ S,
                                            const float* __restrict__ P,
                                            const float* __restrict__ G,
                                            float* __restrict__ H) {
  __shared__ float sP[1024], hT[512], sT[512];
  const int lane = threadIdx.x;
  const int g = blockIdx.x;
  for (int i = lane; i < 1024; i += 32) sP[i] = P[i];
  __syncthreads();
  float2 fP[2][8];
  preload(sP, fP);
  for (int c = 0; c < GC; ++c)
    hT[lane * GC + c] = G[(g * GC + c) * 32 + lane];
  __syncthreads();
  for (int m = 0; m < SUPER; ++m) {
    for (int c = 0; c < GC; ++c) {
      const int chunk = (g * GC + c) * SUPER + m;
      H[chunk * 32 + lane] = hT[lane * GC + c];
      sT[lane * GC + c] = S[chunk * 32 + lane];
    }
    __syncthreads();
    v8f acc[2];
    zacc(acc);
    gemm_acc(fP, hT, acc);
    add_tile(acc, sT);
    __syncthreads();
    store_acc(hT, acc);
    __syncthreads();
  }
}

// --- Phase 3: replay chunks from H_c, emit tanh(C h + D u + n) -------------
__global__ __launch_bounds__(32) void k_phase3(const float* __restrict__ u,
                                               const float* __restrict__ noise,
                                               const float* __restrict__ A,
                                               const float* __restrict__ B,
                                               const float* __restrict__ C,
                                               const float* __restrict__ D,
                                               const float* __restrict__ H,
                                               float* __restrict__ out) {
  __shared__ float sA[1024], sB[1024], sC[1024], sD[1024];
  __shared__ float hT[512], uT[512], yT[512];
  const int lane = threadIdx.x;
  const int g = blockIdx.x;
  for (int i = lane; i < 1024; i += 32) {
    sA[i] = A[i]; sB[i] = B[i]; sC[i] = C[i]; sD[i] = D[i];
  }
  __syncthreads();
  float2 fA[2][8], fB[2][8], fC[2][8], fD[2][8];
  preload(sA, fA);
  preload(sB, fB);
  preload(sC, fC);
  preload(sD, fD);
  for (int c = 0; c < GC; ++c)
    hT[lane * GC + c] = H[(g * GC + c) * 32 + lane];
  __syncthreads();
  const int tBase = g * GC * L;
  for (int j = 0; j < L; ++j) {
    for (int c = 0; c < GC; ++c)
      uT[lane * GC + c] = u[lane * SEQ + tBase + c * L + j];
    __syncthreads();
    v8f ya[2], ha[2];
    zacc(ya);
    gemm_acc(fC, hT, ya);      // C @ h
    gemm_acc(fD, uT, ya);      // + D @ u
    zacc(ha);
    gemm_acc(fA, hT, ha);      // A @ h
    gemm_acc(fB, uT, ha);      // + B @ u
    __syncthreads();
    store_acc(yT, ya);
    store_acc(hT, ha);
    __syncthreads();
    for (int c = 0; c < GC; ++c) {
      const int t = tBase + c * L + j;
      out[lane * SEQ + t] = tanhf(yT[lane * GC + c] + noise[t]);
    }
  }
}

// ---------------------------------------------------------------------------
extern "C" void kernel_launch(void* const* d_in, const int* in_sizes, int n_in,
                              void* d_out, int out_size, void* d_ws, size_t ws_size,
                              hipStream_t stream) {
  (void)in_sizes; (void)n_in; (void)out_size; (void)ws_size;
  const float* u     = (const float*)d_in[0];
  const float* noise = (const float*)d_in[1];
  const float* A     = (const float*)d_in[2];
  const float* B     = (const float*)d_in[3];
  const float* C     = (const float*)d_in[4];
  const float* D     = (const float*)d_in[5];
  float* out = (float*)d_out;

  float* ws = (float*)d_ws;
  float* S  = ws;                        // CHUNKS*32 = 262144
  float* H  = S  + CHUNKS * 32;          // 262144
  float* S2 = H  + CHUNKS * 32;          // NSUP*32 = 4096
  float* G  = S2 + NSUP * 32;            // 4096
  float* P  = G  + NSUP * 32;            // 1024
  float* P2 = P  + 1024;                 // 1024   (~2.1 MB total)

  k_phase1<<<NG,  32, 0, stream>>>(u, A, B, S);
  k_powers<<<1,   32, 0, stream>>>(A, P, P2);
  k_p2a   <<<NSG, 32, 0, stream>>>(S, P, S2);
  k_p2b   <<<1,   32, 0, stream>>>(S2, P2, G);
  k_p2c   <<<NSG, 32, 0, stream>>>(S, P, G, H);
  k_phase3<<<NG,  32, 0, stream>>>(u, noise, A, B, C, D, H, out);
}